// NaivePatSoftmaxRNN_46488726012384
// MI455X (gfx1250) — compile-verified
//
#include <hip/hip_runtime.h>
#include <math.h>

// Sizes fixed by the reference: T=128, B=16, P=256, H=256.
#define T_STEPS 128
#define BATCH   16
#define P_DIM   256
#define H_DIM   256
#define PST     260            // padded LDS row stride in floats (260 mod 64 banks = 4 -> conflict-free column walks)
#define NTHREADS 512
#define NWAVES   16

typedef __attribute__((ext_vector_type(2))) float v2f;
typedef __attribute__((ext_vector_type(8))) float v8f;

__global__ __launch_bounds__(NTHREADS, 1)
void pat_softmax_rnn_kernel(const float* __restrict__ inp,    // [T,B,H]
                            const float* __restrict__ pat0,   // [B,P,H]
                            float* __restrict__ out_h,        // [T,B,H]
                            float* __restrict__ out_p)        // [T,B,P,H]
{
    // Pattern bank resident in LDS for the whole scan: 256*260*4 = 266,240 B (<320 KB/WGP).
    __shared__ float pat_s[P_DIM * PST];
    __shared__ float h_s[H_DIM];
    __shared__ float raw_s[P_DIM];
    __shared__ float resp_s[P_DIM];
    __shared__ float red_s[NWAVES];
    __shared__ float scal_s[2];

    const int b    = blockIdx.x;        // one workgroup per batch chain
    const int tid  = threadIdx.x;
    const int lane = tid & 31;
    const int wid  = tid >> 5;          // 16 waves: wave w owns row/col tile [16w, 16w+16)
    const int m    = lane & 15;         // WMMA fragment row (M) this lane holds
    const int koff = (lane >> 4) * 2;   // lanes 16-31 hold K+2,K+3 per f32 16x16x4 layout
    const int pr   = wid * 16;

    // ---- load initial pattern into LDS ----
    for (int e = tid; e < P_DIM * H_DIM; e += NTHREADS) {
        int p = e >> 8, j = e & (H_DIM - 1);
        pat_s[p * PST + j] = pat0[(size_t)b * P_DIM * H_DIM + e];
    }

    for (int t = 0; t < T_STEPS; ++t) {
        __syncthreads();   // pat_s stable from previous step's update

        // h = input[t, b, :]
        if (tid < H_DIM) h_s[tid] = inp[((size_t)t * BATCH + b) * H_DIM + tid];
        __syncthreads();

        // ---- raw = pat @ h  (V_WMMA_F32_16X16X4_F32, B broadcast across N) ----
        {
            v8f acc = {};
            const float* arow = &pat_s[(pr + m) * PST];
            for (int kk = 0; kk < H_DIM; kk += 4) {
                const int kb = kk + koff;
                v2f a;  a.x = arow[kb];      a.y = arow[kb + 1];
                v2f bb; bb.x = h_s[kb];      bb.y = h_s[kb + 1];
                acc = __builtin_amdgcn_wmma_f32_16x16x4_f32(
                          false, a, false, bb, (short)0, acc, false, false);
            }
            // C layout: VGPR i = row M=i (lanes 0-15) / M=i+8 (lanes 16-31); all N columns equal.
            if (lane == 0) {
                #pragma unroll
                for (int i = 0; i < 8; ++i) raw_s[pr + i] = acc[i];
            } else if (lane == 16) {
                #pragma unroll
                for (int i = 0; i < 8; ++i) raw_s[pr + 8 + i] = acc[i];
            }
        }
        __syncthreads();

        // ---- masked softmax: resp = softmax(where(raw >= 0.9*mx, raw, 0) / mx * 10) ----
        float rv = (tid < P_DIM) ? raw_s[tid] : -INFINITY;
        float vmax = rv;
        for (int off = 16; off > 0; off >>= 1) vmax = fmaxf(vmax, __shfl_xor(vmax, off));
        if (lane == 0) red_s[wid] = vmax;
        __syncthreads();
        if (tid == 0) {
            float mxx = red_s[0];
            for (int i = 1; i < NWAVES; ++i) mxx = fmaxf(mxx, red_s[i]);
            scal_s[0] = mxx;
        }
        __syncthreads();
        const float mx = scal_s[0];
        const float scale = 10.0f / mx;
        float sv = (tid < P_DIM) ? ((rv >= 0.9f * mx) ? rv * scale : 0.0f) : -INFINITY;
        float smax = sv;
        for (int off = 16; off > 0; off >>= 1) smax = fmaxf(smax, __shfl_xor(smax, off));
        if (lane == 0) red_s[wid] = smax;
        __syncthreads();
        if (tid == 0) {
            float m2 = red_s[0];
            for (int i = 1; i < NWAVES; ++i) m2 = fmaxf(m2, red_s[i]);
            scal_s[1] = m2;
        }
        __syncthreads();
        const float m2 = scal_s[1];
        float ev = (tid < P_DIM) ? __expf(sv - m2) : 0.0f;
        float esum = ev;
        for (int off = 16; off > 0; off >>= 1) esum += __shfl_xor(esum, off);
        if (lane == 0) red_s[wid] = esum;
        __syncthreads();
        if (tid == 0) {
            float s = 0.0f;
            for (int i = 0; i < NWAVES; ++i) s += red_s[i];
            scal_s[0] = s;
        }
        __syncthreads();
        if (tid < P_DIM) resp_s[tid] = ev / scal_s[0];
        __syncthreads();

        // ---- new_h = pat^T @ resp  (WMMA on transposed tiles, old pat) ----
        {
            v8f acc = {};
            const int jb = pr;  // this wave's 16-column tile of H
            for (int pp = 0; pp < P_DIM; pp += 4) {
                const int kb = pp + koff;
                v2f a;  a.x = pat_s[kb * PST + jb + m];
                        a.y = pat_s[(kb + 1) * PST + jb + m];
                v2f bb; bb.x = resp_s[kb];  bb.y = resp_s[kb + 1];
                acc = __builtin_amdgcn_wmma_f32_16x16x4_f32(
                          false, a, false, bb, (short)0, acc, false, false);
            }
            float* oh = out_h + ((size_t)t * BATCH + b) * H_DIM + jb;
            if (lane == 0) {
                #pragma unroll
                for (int i = 0; i < 8; ++i) oh[i] = acc[i];
            } else if (lane == 16) {
                #pragma unroll
                for (int i = 0; i < 8; ++i) oh[8 + i] = acc[i];
            }
        }
        __syncthreads();   // all reads of old pat done before in-place update

        // ---- pat = l2_normalize(0.999*pat + resp (x) h); stream rows to all_pats[t] ----
        {
            float* op = out_p + ((size_t)t * BATCH + b) * P_DIM * H_DIM;
            for (int r = 0; r < 16; ++r) {
                const int p = pr + r;
                const float rp = resp_s[p];
                float* row = &pat_s[p * PST];
                float ss = 0.0f;
                #pragma unroll
                for (int jj = 0; jj < H_DIM; jj += 32) {
                    const int j = jj + lane;
                    const float v = 0.999f * row[j] + rp * h_s[j];
                    row[j] = v;
                    ss += v * v;
                }
                for (int off = 16; off > 0; off >>= 1) ss += __shfl_xor(ss, off);
                const float inv = 1.0f / (sqrtf(ss) + 1e-10f);
                #pragma unroll
                for (int jj = 0; jj < H_DIM; jj += 32) {
                    const int j = jj + lane;
                    const float v = row[j] * inv;
                    row[j] = v;
                    op[(size_t)p * H_DIM + j] = v;   // coalesced 128B store per sweep
                }
            }
        }
    }
}

extern "C" void kernel_launch(void* const* d_in, const int* in_sizes, int n_in,
                              void* d_out, int out_size, void* d_ws, size_t ws_size,
                              hipStream_t stream) {
    const float* inp  = (const float*)d_in[0];   // [T,B,H] f32
    const float* pat0 = (const float*)d_in[1];   // [B,P,H] f32
    float* out   = (float*)d_out;
    float* out_h = out;                                          // [T,B,H]
    float* out_p = out + (size_t)T_STEPS * BATCH * H_DIM;        // [T,B,P,H]
    pat_softmax_rnn_kernel<<<BATCH, NTHREADS, 0, stream>>>(inp, pat0, out_h, out_p);
}